// SDFLoss_8967891714450
// MI455X (gfx1250) — compile-verified
//
#include <hip/hip_runtime.h>
#include <math.h>

// Problem constants (from reference)
#define NP    64
#define NV    6890
#define VSUB  256
#define SSTR  26        // NV / VSUB
#define NG    32
#define NG3   32768
#define BAND  0.15f

typedef __attribute__((ext_vector_type(2))) float v2f;
typedef __attribute__((ext_vector_type(8))) float v8f;

// ---------------- workspace layout (float offsets) ----------------
#define WS_BMIN   ((size_t)0)          // 64*3
#define WS_BMAX   ((size_t)192)        // 64*3
#define WS_CENTER ((size_t)384)        // 64*3
#define WS_SCALE  ((size_t)576)        // 64
#define WS_VALID  ((size_t)640)        // 64
#define WS_NV     ((size_t)704)        // 1
#define WS_ACC    ((size_t)705)        // 1
#define WS_BPACK  ((size_t)768)        // 64*256*4 float4-packed B matrices
#define WS_PHI    ((size_t)(768 + 65536)) // 64*32768 phi grids

// ---------------- kernel 1: per-person bbox / center / scale ----------------
__global__ void k_bbox(const float* __restrict__ verts,
                       const float* __restrict__ trans,
                       float* __restrict__ ws) {
    int p = blockIdx.x, t = threadIdx.x;
    float tx = trans[p*3+0], ty = trans[p*3+1], tz = trans[p*3+2];
    float mnx = 1e30f, mny = 1e30f, mnz = 1e30f;
    float mxx = -1e30f, mxy = -1e30f, mxz = -1e30f;
    for (int v = t; v < NV; v += 256) {
        const float* q = verts + ((size_t)p*NV + v)*3;
        float x = q[0]+tx, y = q[1]+ty, z = q[2]+tz;
        mnx = fminf(mnx,x); mny = fminf(mny,y); mnz = fminf(mnz,z);
        mxx = fmaxf(mxx,x); mxy = fmaxf(mxy,y); mxz = fmaxf(mxz,z);
    }
    __shared__ float s[6][256];
    s[0][t]=mnx; s[1][t]=mny; s[2][t]=mnz;
    s[3][t]=mxx; s[4][t]=mxy; s[5][t]=mxz;
    __syncthreads();
    for (int off = 128; off > 0; off >>= 1) {
        if (t < off) {
            s[0][t]=fminf(s[0][t],s[0][t+off]);
            s[1][t]=fminf(s[1][t],s[1][t+off]);
            s[2][t]=fminf(s[2][t],s[2][t+off]);
            s[3][t]=fmaxf(s[3][t],s[3][t+off]);
            s[4][t]=fmaxf(s[4][t],s[4][t+off]);
            s[5][t]=fmaxf(s[5][t],s[5][t+off]);
        }
        __syncthreads();
    }
    if (t == 0) {
        float bmnx=s[0][0], bmny=s[1][0], bmnz=s[2][0];
        float bmxx=s[3][0], bmxy=s[4][0], bmxz=s[5][0];
        ws[WS_BMIN+p*3+0]=bmnx; ws[WS_BMIN+p*3+1]=bmny; ws[WS_BMIN+p*3+2]=bmnz;
        ws[WS_BMAX+p*3+0]=bmxx; ws[WS_BMAX+p*3+1]=bmxy; ws[WS_BMAX+p*3+2]=bmxz;
        ws[WS_CENTER+p*3+0]=0.5f*(bmnx+bmxx);
        ws[WS_CENTER+p*3+1]=0.5f*(bmny+bmxy);
        ws[WS_CENTER+p*3+2]=0.5f*(bmnz+bmxz);
        float ext = fmaxf(bmxx-bmnx, fmaxf(bmxy-bmny, bmxz-bmnz));
        ws[WS_SCALE+p] = 0.6f * ext;   // (1+0.2)*0.5*maxExtent
    }
}

// ---------------- kernel 2: validity mask + nv, zero accumulator ----------------
__global__ void k_valid(float* __restrict__ ws) {
    int i = threadIdx.x;   // 64 threads
    __shared__ float sb[384];
    for (int k = i; k < 384; k += 64) sb[k] = ws[WS_BMIN + k]; // bmin(192) then bmax(192)
    __syncthreads();
    const float* bmin = sb;
    const float* bmax = sb + 192;
    bool ok = true;
    for (int j = 0; j < NP; ++j) {
        bool ov = true;
        #pragma unroll
        for (int a = 0; a < 3; ++a)
            ov = ov && (bmin[i*3+a] <= bmax[j*3+a]) && (bmin[j*3+a] <= bmax[i*3+a]);
        ok = ok && ov;
    }
    ws[WS_VALID + i] = ok ? 1.0f : 0.0f;
    __shared__ float cnt[64];
    cnt[i] = ok ? 1.0f : 0.0f;
    __syncthreads();
    for (int off = 32; off > 0; off >>= 1) {
        if (i < off) cnt[i] += cnt[i+off];
        __syncthreads();
    }
    if (i == 0) { ws[WS_NV] = cnt[0]; ws[WS_ACC] = 0.0f; }
}

// ---------------- kernel 3: pack per-person WMMA B matrices ----------------
// B column for subsampled vertex n: [-2x, -2y, -2z, |v|^2] in person's cs frame
__global__ void k_buildB(const float* __restrict__ verts,
                         const float* __restrict__ trans,
                         float* __restrict__ ws) {
    int p = blockIdx.x, t = threadIdx.x;     // t = subsample index 0..255
    float cx = ws[WS_CENTER+p*3+0], cy = ws[WS_CENTER+p*3+1], cz = ws[WS_CENTER+p*3+2];
    float inv = 1.0f / ws[WS_SCALE+p];
    int v = t * SSTR;
    const float* q = verts + ((size_t)p*NV + v)*3;
    float x = (q[0] + trans[p*3+0] - cx) * inv;
    float y = (q[1] + trans[p*3+1] - cy) * inv;
    float z = (q[2] + trans[p*3+2] - cz) * inv;
    float4 b = make_float4(-2.0f*x, -2.0f*y, -2.0f*z, x*x + y*y + z*z);
    ((float4*)(ws + WS_BPACK))[p*VSUB + t] = b;
}

// ---------------- kernel 4: phi grid via V_WMMA_F32_16X16X4_F32 ----------------
// B-pack staged memory->LDS with GLOBAL_LOAD_ASYNC_TO_LDS_B128 (ASYNCcnt path),
// then one wave handles 16 grid cells: 16 WMMAs (16 verts each) -> running min
// -> cross-lane min -> phi = relu(band - sqrt(min + |g|^2)).
__global__ void k_phi(float* __restrict__ ws) {
    int p  = blockIdx.x >> 8;          // 256 blocks per person
    int bt = blockIdx.x & 255;
    int t = threadIdx.x, lane = t & 31, w = t >> 5;

    __shared__ float4 sB[VSUB];

    {
        // Async DMA: each thread copies one float4 row of the B-pack directly
        // into LDS (no VGPR data round-trip). Tracked by ASYNCcnt.
        //
        // The generic address of __shared__ memory has the wave's LDS byte
        // offset in its low 32 bits,
        // which is exactly what the VDST operand of
        // GLOBAL_LOAD_ASYNC_TO_LDS_B128 expects. The ptrtoint also captures
        // sB, so the "memory"-clobbering asm is a legal writer of it and the
        // later ds_load reads cannot be folded away.
        unsigned lds_addr = (unsigned)(uintptr_t)(&sB[t]);
        const float4* gsrc = ((const float4*)(ws + WS_BPACK)) + p*VSUB + t;
        unsigned long long gaddr = (unsigned long long)(uintptr_t)gsrc;
        asm volatile("global_load_async_to_lds_b128 %0, %1, off"
                     :
                     : "v"(lds_addr), "v"(gaddr)
                     : "memory");
        asm volatile("s_wait_asynccnt 0x0" ::: "memory");
    }
    __syncthreads();

    int tileBase = bt*128 + w*16;      // 8 waves * 16 cells = 128 cells per block
    int M = lane & 15;
    bool hi = lane >= 16;

    // A row for cell tileBase+M: [gx, gy, gz, 1]
    // f32 16x4 A layout: lanes 0-15 -> K=0,1 ; lanes 16-31 -> K=2,3
    int cell = tileBase + M;
    int cx = cell & 31, cy = (cell >> 5) & 31, cz = cell >> 10;
    float gx = (2.0f*cx + 1.0f)*(1.0f/32.0f) - 1.0f;
    float gy = (2.0f*cy + 1.0f)*(1.0f/32.0f) - 1.0f;
    float gz = (2.0f*cz + 1.0f)*(1.0f/32.0f) - 1.0f;
    v2f A;
    A[0] = hi ? gz : gx;
    A[1] = hi ? 1.0f : gy;

    v8f czero = {};
    v8f mn;
    #pragma unroll
    for (int k = 0; k < 16; ++k) {
        float4 bv = sB[k*16 + M];      // vertex n = k*16 + (lane&15)
        v2f B;
        B[0] = hi ? bv.z : bv.x;       // K=0 -> -2x ; K=2 -> -2z
        B[1] = hi ? bv.w : bv.y;       // K=1 -> -2y ; K=3 -> |v|^2
        // D[M][N] = |v_N|^2 - 2 g_M . v_N
        v8f d = __builtin_amdgcn_wmma_f32_16x16x4_f32(
            false, A, false, B, (short)0, czero, false, false);
        if (k == 0) {
            mn = d;
        } else {
            #pragma unroll
            for (int i = 0; i < 8; ++i) mn[i] = fminf(mn[i], d[i]);
        }
    }

    // min across the 16 lanes of each half (masks 1,2,4,8 stay within halves)
    #pragma unroll
    for (int i = 0; i < 8; ++i) {
        float x = mn[i];
        x = fminf(x, __shfl_xor(x, 1, 32));
        x = fminf(x, __shfl_xor(x, 2, 32));
        x = fminf(x, __shfl_xor(x, 4, 32));
        x = fminf(x, __shfl_xor(x, 8, 32));
        mn[i] = x;
    }

    // lanes 0-15 hold mins for M=0..7 (VGPR r -> M=r); lanes 16-31 for M=8..15
    if (M == 0) {                       // lane 0 and lane 16 write
        int mBase = hi ? 8 : 0;
        float* phi = ws + WS_PHI + (size_t)p * NG3;
        #pragma unroll
        for (int r = 0; r < 8; ++r) {
            int c2 = tileBase + mBase + r;
            int x = c2 & 31, y = (c2 >> 5) & 31, z = c2 >> 10;
            float hx = (2.0f*x + 1.0f)*(1.0f/32.0f) - 1.0f;
            float hy = (2.0f*y + 1.0f)*(1.0f/32.0f) - 1.0f;
            float hz = (2.0f*z + 1.0f)*(1.0f/32.0f) - 1.0f;
            float g2 = hx*hx + hy*hy + hz*hz;
            float d2 = fmaxf(mn[r] + g2, 0.0f);
            phi[c2] = fmaxf(BAND - sqrtf(d2), 0.0f);
        }
    }
}

// ---------------- kernel 5: trilinear sample + pair accumulation ----------------
__device__ inline float tri_corner(const float* __restrict__ vol,
                                   int xi, int yi, int zi, float w) {
    if ((unsigned)xi < 32u && (unsigned)yi < 32u && (unsigned)zi < 32u)
        return vol[(zi << 10) + (yi << 5) + xi] * w;
    return 0.0f;
}

__global__ void k_sample(const float* __restrict__ verts,
                         const float* __restrict__ trans,
                         float* __restrict__ ws) {
    int pair = blockIdx.x;
    int i = pair >> 6, j = pair & 63;
    if (i == j) return;
    if (ws[WS_VALID + i] == 0.0f || ws[WS_VALID + j] == 0.0f) return;

    float cx = ws[WS_CENTER+i*3+0], cy = ws[WS_CENTER+i*3+1], cz = ws[WS_CENTER+i*3+2];
    float inv = 1.0f / ws[WS_SCALE+i];
    float tx = trans[j*3+0], ty = trans[j*3+1], tz = trans[j*3+2];
    const float* phi = ws + WS_PHI + (size_t)i * NG3;

    float acc = 0.0f;
    for (int v = threadIdx.x; v < NV; v += 256) {
        const float* q = verts + ((size_t)j*NV + v)*3;
        float lx = (q[0]+tx-cx)*inv, ly = (q[1]+ty-cy)*inv, lz = (q[2]+tz-cz)*inv;
        // grid_sample align_corners=False: x = ((p+1)*32 - 1)/2 = 16p + 15.5
        float x = lx*16.0f + 15.5f, y = ly*16.0f + 15.5f, z = lz*16.0f + 15.5f;
        float fx0 = floorf(x), fy0 = floorf(y), fz0 = floorf(z);
        int x0 = (int)fx0, y0 = (int)fy0, z0 = (int)fz0;
        float fx = x - fx0, fy = y - fy0, fz = z - fz0;
        float gx0 = 1.0f - fx, gy0 = 1.0f - fy, gz0 = 1.0f - fz;
        acc += tri_corner(phi, x0,   y0,   z0,   gx0*gy0*gz0)
             + tri_corner(phi, x0+1, y0,   z0,   fx *gy0*gz0)
             + tri_corner(phi, x0,   y0+1, z0,   gx0*fy *gz0)
             + tri_corner(phi, x0+1, y0+1, z0,   fx *fy *gz0)
             + tri_corner(phi, x0,   y0,   z0+1, gx0*gy0*fz)
             + tri_corner(phi, x0+1, y0,   z0+1, fx *gy0*fz)
             + tri_corner(phi, x0,   y0+1, z0+1, gx0*fy *fz)
             + tri_corner(phi, x0+1, y0+1, z0+1, fx *fy *fz);
    }
    __shared__ float red[256];
    red[threadIdx.x] = acc;
    __syncthreads();
    for (int off = 128; off > 0; off >>= 1) {
        if (threadIdx.x < off) red[threadIdx.x] += red[threadIdx.x + off];
        __syncthreads();
    }
    if (threadIdx.x == 0) atomicAdd(ws + WS_ACC, red[0]);
}

// ---------------- kernel 6: finalize ----------------
__global__ void k_final(const float* __restrict__ ws, float* __restrict__ out) {
    float nv = fmaxf(ws[WS_NV], 1.0f);
    out[0] = ws[WS_ACC] / (nv * nv);
}

// ---------------- launch ----------------
extern "C" void kernel_launch(void* const* d_in, const int* in_sizes, int n_in,
                              void* d_out, int out_size, void* d_ws, size_t ws_size,
                              hipStream_t stream) {
    (void)in_sizes; (void)n_in; (void)out_size; (void)ws_size;
    const float* verts = (const float*)d_in[0];   // [64,6890,3] f32
    const float* trans = (const float*)d_in[1];   // [64,3] f32
    // d_in[2] = faces (unused by the reference computation)
    float* out = (float*)d_out;
    float* ws  = (float*)d_ws;

    k_bbox  <<<NP,        256, 0, stream>>>(verts, trans, ws);
    k_valid <<<1,         64,  0, stream>>>(ws);
    k_buildB<<<NP,        256, 0, stream>>>(verts, trans, ws);
    k_phi   <<<NP * 256,  256, 0, stream>>>(ws);      // 256 blocks/person, 128 cells/block
    k_sample<<<NP * NP,   256, 0, stream>>>(verts, trans, ws);
    k_final <<<1,         1,   0, stream>>>(ws, out);
}